// HeteroConvLayer_74045236183058
// MI455X (gfx1250) — compile-verified
//
#include <hip/hip_runtime.h>

// MI455X / gfx1250, wave32. fp32 WMMA 16x16x4 for full-precision GEMMs;
// LDS-staged weights; f32 global atomics for segment-mean; shfl_xor row norm.

typedef __attribute__((ext_vector_type(2))) float v2f;
typedef __attribute__((ext_vector_type(8))) float v8f;

#define LDS_PITCH 132   // 132 % 64 == 4 -> conflict-free ds_load_b64; even -> 8B aligned

__global__ void hsage_zero_kernel(float* __restrict__ p, int n) {
    int i = blockIdx.x * blockDim.x + threadIdx.x;
    if (i < n) p[i] = 0.0f;
}

// One thread per (edge, feature). A wave32 covers 32 consecutive features of one
// edge -> coalesced 128B row segments for both the gather and the atomic scatter.
__global__ void hsage_scatter_kernel(const float* __restrict__ xsrc,
                                     const int* __restrict__ si,
                                     const int* __restrict__ di,
                                     float* __restrict__ agg,
                                     float* __restrict__ cnt,
                                     int E) {
    int tid = blockIdx.x * blockDim.x + threadIdx.x;
    int e = tid >> 7;
    int f = tid & 127;
    if (e >= E) return;
    int s = si[e];
    int d = di[e];
    atomicAdd(&agg[d * 128 + f], xsrc[s * 128 + f]);
    if (f == 0) atomicAdd(&cnt[d], 1.0f);
}

// Fused: out[row] (=|+=) normalize( X[row]·WL^T + b + (agg[row]/cnt[row])·WR^T )
// One wave per 16-row tile; 8 N-tiles of 16 cols; K=128 via 32 wmma_f32_16x16x4 steps.
__global__ __launch_bounds__(256)
void hsage_gemm_kernel(const float* __restrict__ xdst,
                       const float* __restrict__ agg,
                       const float* __restrict__ cnt,
                       const float* __restrict__ wl,    // [128 out][128 in] row-major
                       const float* __restrict__ bias,  // [128]
                       const float* __restrict__ wr,    // [128 out][128 in]
                       float* __restrict__ out,         // pre-offset dst region
                       int nDst, int accumulate) {
    extern __shared__ float lds[];
    float* lwl = lds;
    float* lwr = lds + 128 * LDS_PITCH;

    // Stage both weight matrices into LDS once per block (coalesced).
    for (int idx = threadIdx.x; idx < 128 * 128; idx += blockDim.x) {
        int j = idx >> 7, k = idx & 127;
        lwl[j * LDS_PITCH + k] = wl[idx];
        lwr[j * LDS_PITCH + k] = wr[idx];
    }
    __syncthreads();

    const int lane = threadIdx.x & 31;
    const int wave = threadIdx.x >> 5;
    const int half = lane >> 4;     // K/M half selector per WMMA layouts
    const int l16  = lane & 15;
    const int numTiles = nDst >> 4; // nDst is a multiple of 16 for all edge types

    for (int bt = blockIdx.x; bt * 8 < numTiles; bt += gridDim.x) {
        int tile = bt * 8 + wave;
        if (tile >= numTiles) continue;
        int row0 = tile << 4;

        // ---- A regs: X tile, WMMA f32 A layout (lane: row=l16, K pair at 4s+2*half)
        v2f ax[32];
        {
            const float* p = xdst + (row0 + l16) * 128 + 2 * half;
            #pragma unroll
            for (int s = 0; s < 32; ++s)
                ax[s] = *(const v2f*)(p + 4 * s);
        }

        // ---- C init with bias (broadcast down each column)
        v8f cacc[8];
        #pragma unroll
        for (int jt = 0; jt < 8; ++jt) {
            float bv = bias[jt * 16 + l16];
            v8f c;
            #pragma unroll
            for (int v = 0; v < 8; ++v) c[v] = bv;
            cacc[jt] = c;
        }

        // ---- pass 1: C += X · WL^T   (B[k][n] = WL[n][k], read from LDS)
        #pragma unroll
        for (int jt = 0; jt < 8; ++jt) {
            const float* bp = lwl + (jt * 16 + l16) * LDS_PITCH + 2 * half;
            v8f c = cacc[jt];
            #pragma unroll
            for (int s = 0; s < 32; ++s) {
                v2f b = *(const v2f*)(bp + 4 * s);
                c = __builtin_amdgcn_wmma_f32_16x16x4_f32(
                        false, ax[s], false, b, (short)0, c, false, false);
            }
            cacc[jt] = c;
        }

        // ---- A regs: mean-aggregated tile (fold 1/cnt into A)
        {
            float cv = cnt[row0 + l16];
            float invc = 1.0f / fmaxf(cv, 1.0f);
            const float* p = agg + (row0 + l16) * 128 + 2 * half;
            #pragma unroll
            for (int s = 0; s < 32; ++s) {
                v2f t = *(const v2f*)(p + 4 * s);
                t[0] *= invc;
                t[1] *= invc;
                ax[s] = t;
            }
        }

        // ---- pass 2: C += AGG · WR^T
        #pragma unroll
        for (int jt = 0; jt < 8; ++jt) {
            const float* bp = lwr + (jt * 16 + l16) * LDS_PITCH + 2 * half;
            v8f c = cacc[jt];
            #pragma unroll
            for (int s = 0; s < 32; ++s) {
                v2f b = *(const v2f*)(bp + 4 * s);
                c = __builtin_amdgcn_wmma_f32_16x16x4_f32(
                        false, ax[s], false, b, (short)0, c, false, false);
            }
            cacc[jt] = c;
        }

        // ---- per-row L2 norm. Element (lane,v,jt) = D[v+8*half][jt*16+l16].
        float ss[8];
        #pragma unroll
        for (int v = 0; v < 8; ++v) {
            float a = 0.0f;
            #pragma unroll
            for (int jt = 0; jt < 8; ++jt) { float x = cacc[jt][v]; a += x * x; }
            ss[v] = a;
        }
        #pragma unroll
        for (int v = 0; v < 8; ++v) {
            float a = ss[v];
            a += __shfl_xor(a, 1, 32);   // butterfly within the 16-lane half
            a += __shfl_xor(a, 2, 32);
            a += __shfl_xor(a, 4, 32);
            a += __shfl_xor(a, 8, 32);
            ss[v] = 1.0f / fmaxf(sqrtf(a), 1e-12f);
        }
        #pragma unroll
        for (int jt = 0; jt < 8; ++jt) {
            #pragma unroll
            for (int v = 0; v < 8; ++v) cacc[jt][v] *= ss[v];
        }

        // ---- write (first sage for a region) or read-add (second sage)
        #pragma unroll
        for (int jt = 0; jt < 8; ++jt) {
            #pragma unroll
            for (int v = 0; v < 8; ++v) {
                int oidx = (row0 + v + 8 * half) * 128 + jt * 16 + l16;
                if (accumulate) out[oidx] += cacc[jt][v];
                else            out[oidx]  = cacc[jt][v];
            }
        }
    }
}

extern "C" void kernel_launch(void* const* d_in, const int* in_sizes, int n_in,
                              void* d_out, int out_size, void* d_ws, size_t ws_size,
                              hipStream_t stream) {
    const float* xa = (const float*)d_in[0];
    const float* xe = (const float*)d_in[1];
    const float* xf = (const float*)d_in[2];
    const int NA = in_sizes[0] / 128;
    const int NE = in_sizes[1] / 128;
    const int NF = in_sizes[2] / 128;
    const int E  = in_sizes[3];

    const int*   si[5];
    const int*   di[5];
    const float* wl[5];
    const float* bb[5];
    const float* wr[5];
    for (int i = 0; i < 5; ++i) {
        si[i] = (const int*)d_in[3 + 2 * i];
        di[i] = (const int*)d_in[4 + 2 * i];
        wl[i] = (const float*)d_in[13 + 3 * i];
        bb[i] = (const float*)d_in[14 + 3 * i];
        wr[i] = (const float*)d_in[15 + 3 * i];
    }

    float* out = (float*)d_out;
    // Edge types: (art->ent), (art->fact), (ent->art), (ent->fact), (fact->ent)
    const float* srcs[5] = { xa, xa, xe, xe, xf };
    const float* dsts[5] = { xe, xf, xa, xf, xe };
    const int    ndst[5] = { NE, NF, NA, NF, NE };
    long long    ooff[5] = { (long long)NA * 128, (long long)(NA + NE) * 128, 0,
                             (long long)(NA + NE) * 128, (long long)NA * 128 };
    const int    accm[5] = { 0, 0, 0, 1, 1 };   // order guarantees store-before-add

    float* agg = (float*)d_ws;              // max NE*128 floats (~102 MB)
    float* cnt = agg + (size_t)NE * 128;    // max NE floats

    const size_t ldsBytes = (size_t)2 * 128 * LDS_PITCH * sizeof(float); // 132 KB < 320 KB/WGP

    for (int t = 0; t < 5; ++t) {
        int nd = ndst[t];

        int nz = nd * 128;
        hsage_zero_kernel<<<(nz + 255) / 256, 256, 0, stream>>>(agg, nz);
        hsage_zero_kernel<<<(nd + 255) / 256, 256, 0, stream>>>(cnt, nd);

        int sThreads = E * 128;
        hsage_scatter_kernel<<<(sThreads + 255) / 256, 256, 0, stream>>>(
            srcs[t], si[t], di[t], agg, cnt, E);

        int numTiles = nd >> 4;
        int gBlocks = (numTiles + 7) / 8;
        if (gBlocks > 1024) gBlocks = 1024;
        hsage_gemm_kernel<<<gBlocks, 256, ldsBytes, stream>>>(
            dsts[t], agg, cnt, wl[t], bb[t], wr[t], out + ooff[t], nd, accm[t]);
    }
}